// EfficientSoftNPLoss_17892833755271
// MI455X (gfx1250) — compile-verified
//
#include <hip/hip_runtime.h>
#include <hip/hip_bf16.h>

typedef float v2f __attribute__((ext_vector_type(2)));
typedef float v8f __attribute__((ext_vector_type(8)));
typedef int   v4i __attribute__((ext_vector_type(4)));

#define KNN   30      // K
#define NCAND 60      // 2K candidates per batch element
#define NPAD  64      // padded to 4 WMMA tiles of 16
#define DIM   256     // latent dim
#define TPB   128     // 4 wave32 per block, one block per batch element

#if __has_builtin(__builtin_amdgcn_global_load_async_to_lds_b128)
#define USE_ASYNC_LDS 1
typedef __attribute__((address_space(1))) v4i* g_v4i_p;   // global (AS1)
typedef __attribute__((address_space(3))) v4i* l_v4i_p;   // LDS (AS3)
#else
#define USE_ASYNC_LDS 0
#endif

__device__ __forceinline__ void wait_async_zero() {
#if USE_ASYNC_LDS
#if __has_builtin(__builtin_amdgcn_s_wait_asynccnt)
    __builtin_amdgcn_s_wait_asynccnt(0);
#else
    asm volatile("s_wait_asynccnt 0x0" ::: "memory");
#endif
#endif
}

__global__ void softnp_zero(float* out) { out[0] = 0.0f; }

__global__ __launch_bounds__(TPB)
void softnp_kernel(const float* __restrict__ z_all,
                   const int*   __restrict__ knn,
                   const int*   __restrict__ cells,
                   const int*   __restrict__ negs,
                   float*       __restrict__ out,
                   int batch)
{
    // LDS: row 0 = anchor z_i, rows 1..64 = candidates (60 real + 4 pad)
    __shared__ int   gidx[NPAD + 1];
    __shared__ float mat[(NPAD + 1) * DIM];   // 66560 B
    __shared__ float dist2[NPAD];

    const int tid = threadIdx.x;
    const int b   = blockIdx.x;
    const int ci  = cells[b];

    // ---- Phase 0: resolve gather indices ----
    if (tid < NPAD + 1) {
        int gi;
        if (tid == 0) {
            gi = ci;                                   // anchor
        } else {
            const int j = tid - 1;
            if (j < KNN)        gi = knn[(size_t)ci * KNN + j];
            else if (j < NCAND) gi = negs[(size_t)b * KNN + (j - KNN)];
            else                gi = ci;               // pad -> diff 0, ignored
        }
        gidx[tid] = gi;
    }
    __syncthreads();

    // ---- Phase 1: stage 65 rows (1 KB each) into LDS, 16B per lane/iter.
    // CDNA5 path: GLOBAL_LOAD_ASYNC_TO_LDS_B128 (ASYNCcnt-tracked, no VGPR
    // round-trip); fallback: coalesced global_load_b128 + ds_store_b128.
    for (int i = tid; i < (NPAD + 1) * (DIM / 4); i += TPB) {
        const int r = i >> 6;            // row (64 float4 chunks per row)
        const int c = i & 63;
        const float* src = z_all + (size_t)gidx[r] * DIM + (size_t)c * 4;
        float*       dst = mat + (size_t)r * DIM + (size_t)c * 4;
#if USE_ASYNC_LDS
        __builtin_amdgcn_global_load_async_to_lds_b128(
            (g_v4i_p)(v4i*)src, (l_v4i_p)(v4i*)dst, 0, 0);
#else
        reinterpret_cast<float4*>(dst)[0] =
            reinterpret_cast<const float4*>(src)[0];
#endif
    }
    wait_async_zero();
    __syncthreads();

    // ---- Phase 2: squared distances as the diagonal of WMMA Gram matrices.
    // Wave w handles candidates [16w, 16w+16). For V_WMMA_F32_16X16X4_F32 the
    // A fragment (lane L, vgpr v) = A[M=L%16, K=2*(L/16)+v] and the B fragment
    // (lane L, vgpr v) = B[K=v+2*(L/16), N=L%16] have identical per-lane
    // indexing, so feeding the same float2 of diff data as A and B accumulates
    // C[m,n] = dot(diff_m, diff_n); we keep only the diagonal C[m,m]=||diff_m||^2.
    const int lane = tid & 31;
    const int w    = tid >> 5;           // wave id 0..3 -> tile
    const int half = lane >> 4;          // 0 or 1
    const int mloc = lane & 15;          // candidate-in-tile

    const float* __restrict__ rp = mat + (size_t)(1 + w * 16 + mloc) * DIM;
    const float* __restrict__ xp = mat; // anchor row

    v8f acc[4] = {};                     // 4 chains to hide WMMA RAW latency
    #pragma unroll 2
    for (int c = 0; c < DIM / 4; c += 4) {
        #pragma unroll
        for (int q = 0; q < 4; ++q) {
            const int e = (c + q) * 4 + half * 2;        // even -> 8B aligned
            const v2f cv = *reinterpret_cast<const v2f*>(rp + e);
            const v2f xv = *reinterpret_cast<const v2f*>(xp + e);
            const v2f a  = cv - xv;
            acc[q] = __builtin_amdgcn_wmma_f32_16x16x4_f32(
                false, a, false, a, (short)0, acc[q], false, false);
        }
    }
    v8f s = acc[0] + acc[1] + acc[2] + acc[3];

    // Diagonal extraction per C layout: vgpr v holds M=v (lanes 0-15) and
    // M=v+8 (lanes 16-31, N=lane-16). C[v,v] lives in lane v; C[v+8,v+8] in
    // lane v+24.
    float dval = 0.0f;
    #pragma unroll
    for (int v = 0; v < 8; ++v) {
        if (lane == v)      dval = s[v];
        if (lane == v + 24) dval = s[v];
    }
    if (lane < 8)   dist2[w * 16 + lane]              = dval;
    if (lane >= 24) dist2[w * 16 + 8 + (lane - 24)]   = dval;
    __syncthreads();

    // ---- Phase 3: softmax over 60 dists + loss (wave 0 only) ----
    if (w == 0) {
        float e1 = 0.0f, e2 = 0.0f, n1 = 0.0f;
        {
            const int j = lane;                          // j in [0,32)
            const float d = sqrtf(fmaxf(dist2[j], 0.0f));
            const float e = __expf(-d);                  // TEMPERATURE = 1
            e1 = e;                                      // all j<32 are valid
            if (j < KNN) n1 = e;                         // numerator (kNN part)
        }
        {
            const int j = lane + 32;                     // j in [32,64)
            if (j < NCAND) {
                const float d = sqrtf(fmaxf(dist2[j], 0.0f));
                e2 = __expf(-d);                         // never in numerator
            }
        }
        float se = e1 + e2;
        float sn = n1;
        #pragma unroll
        for (int off = 16; off > 0; off >>= 1) {
            se += __shfl_xor(se, off, 32);
            sn += __shfl_xor(sn, off, 32);
        }
        if (lane == 0) {
            const float loss = -__logf(sn / se + 1e-8f);
            atomicAdd(out, loss * (1.0f / (float)batch));
        }
    }
}

extern "C" void kernel_launch(void* const* d_in, const int* in_sizes, int n_in,
                              void* d_out, int out_size, void* d_ws, size_t ws_size,
                              hipStream_t stream) {
    const float* z_all = (const float*)d_in[0];
    const int*   knn   = (const int*)  d_in[1];
    const int*   cells = (const int*)  d_in[2];
    const int*   negs  = (const int*)  d_in[3];
    float*       out   = (float*)      d_out;

    const int batch = in_sizes[2];     // cell_indices length = BATCH

    softnp_zero<<<1, 1, 0, stream>>>(out);
    softnp_kernel<<<batch, TPB, 0, stream>>>(z_all, knn, cells, negs, out, batch);
}